// MultiResEmbedding_60739427500168
// MI455X (gfx1250) — compile-verified
//
#include <hip/hip_runtime.h>
#include <stdint.h>

// ---------------------------------------------------------------------------
// MultiResEmbedding: out[p,:] = sum_{r=0..5} sum_{c=0..2} W_r[idx(r,c,p), :]
// Memory-bound (≈268 MB HBM). CDNA5 paths:
//   * TDM tensor_load_to_lds stages the 3 small tables (88 KB) per workgroup
//   * per-wave LDS index buffer: offsets computed 1 lane/point, re-read as
//     broadcast ds_load_b64 (same-wave LDS is in-order -> no barriers)
//   * ds_load_b128 LDS gathers (bank-conflict-free), global_load_b128 for the
//     L2-resident big tables, non-temporal b128 streaming store
// ---------------------------------------------------------------------------

#define THREADS 256
#define WAVES_PER_BLOCK (THREADS / 32)
#define CHUNK 32  // points per wave-chunk (one lane computes one point's idx)

static constexpr int kRes[6]    = {16, 32, 64, 128, 256, 512};
// staged tables r=0..2, rows = 3*(res+1): 51, 99, 195  (x 64 floats each)
static constexpr int kLdsOff[3] = {0, 51 * 64, (51 + 99) * 64};
static constexpr int kLdsFloats = (51 + 99 + 195) * 64;  // 22080 fl = 88320 B

typedef uint32_t uv4 __attribute__((ext_vector_type(4)));
typedef int      iv2 __attribute__((ext_vector_type(2)));
typedef int      iv4 __attribute__((ext_vector_type(4)));
typedef int      iv8 __attribute__((ext_vector_type(8)));
typedef float    fv4 __attribute__((ext_vector_type(4)));

struct TrueT  { static constexpr bool value = true;  };
struct FalseT { static constexpr bool value = false; };

// One 1-D TDM transfer: nElems f32 from global -> LDS byte address.
__device__ __forceinline__ void tdm_load_1d(const float* gsrc,
                                            uint32_t ldsByteAddr,
                                            uint32_t nElems) {
  uint64_t ga = (uint64_t)(uintptr_t)gsrc;
  uv4 g0;
  g0[0] = 1u;                                  // count=1 (valid), user mode
  g0[1] = ldsByteAddr;                         // lds_addr (bytes)
  g0[2] = (uint32_t)ga;                        // global_addr[31:0]
  g0[3] = (uint32_t)((ga >> 32) & 0x1FFFFFFu)  // global_addr[56:32]
        | 0x80000000u;                         // type=2 ("image")
  iv8 g1;
  g1[0] = (int)(2u << 16);                     // data_size=2 (4B), no mcast
  g1[1] = (int)((nElems & 0xFFFFu) << 16);     // tensor_dim0[15:0]
  g1[2] = (int)((nElems >> 16) | (1u << 16));  // tensor_dim0 hi | dim1=1
  g1[3] = (int)(nElems << 16);                 // tile_dim0 = nElems (1-D)
  g1[4] = 0;                                   // tile_dim1=0, tile_dim2=0
  g1[5] = (int)nElems;                         // tensor_dim0_stride
  g1[6] = 0;
  g1[7] = 0;
  iv4 z4 = {0, 0, 0, 0};
#if __has_builtin(__builtin_amdgcn_tensor_load_to_lds)
#  if __has_include(<hip/amd_detail/amd_gfx1250_TDM.h>)
  iv8 z8 = {0, 0, 0, 0, 0, 0, 0, 0};
  __builtin_amdgcn_tensor_load_to_lds(g0, g1, z4, z4, z8, 0);  // clang-23 form
#  else
  __builtin_amdgcn_tensor_load_to_lds(g0, g1, z4, z4, 0);      // ROCm 7.2 form
#  endif
#else
  (void)z4;
  asm volatile("tensor_load_to_lds %0, %1" ::"s"(g0), "s"(g1) : "memory");
#endif
}

__global__ __launch_bounds__(THREADS) void MultiResEmbedding_kernel(
    const float* __restrict__ x,
    const float* __restrict__ W0, const float* __restrict__ W1,
    const float* __restrict__ W2, const float* __restrict__ W3,
    const float* __restrict__ W4, const float* __restrict__ W5,
    float* __restrict__ out, int npts) {
  __shared__ __attribute__((aligned(16))) float smem[kLdsFloats];
  __shared__ __attribute__((aligned(16))) int idxBuf[WAVES_PER_BLOCK][CHUNK][18];

  // ---- Stage W0..W2 into LDS via TDM (wave 0 issues; EXEC ignored by TDM) --
  if (threadIdx.x < 32u) {
    uint32_t base = (uint32_t)(uintptr_t)(void*)&smem[0];  // low 32b = LDS off
    tdm_load_1d(W0, base + (uint32_t)kLdsOff[0] * 4u, 51u * 64u);
    tdm_load_1d(W1, base + (uint32_t)kLdsOff[1] * 4u, 99u * 64u);
    tdm_load_1d(W2, base + (uint32_t)kLdsOff[2] * 4u, 195u * 64u);
    __builtin_amdgcn_s_wait_tensorcnt(0);
  }
  __syncthreads();

  const int lane    = (int)(threadIdx.x & 31u);
  const int laneSel = lane & 16;         // 0 for group0, 16 for group1
  const int l16b    = (lane & 15) << 4;  // byte offset of this lane's fv4
  const int waveId  = (int)(threadIdx.x >> 5);
  const int chunkStride = (int)gridDim.x * WAVES_PER_BLOCK * CHUNK;

  const char* Wg[3] = {(const char*)W3, (const char*)W4, (const char*)W5};
  // per-lane base into this wave's index records, at slot `laneSel`
  const int* rec = &idxBuf[waveId][laneSel][0];

  auto runChunk = [&](int pb, auto gtag) {
    constexpr bool GUARD = decltype(gtag)::value;

    // ---- Phase A: one lane per point computes all 18 byte-offsets --------
    int pcl = pb + lane;
    if (GUARD && pcl >= npts) pcl = npts - 1;  // clamp, keep EXEC full
    const float* xp = x + 3 * (size_t)pcl;
    float xv[3];
    xv[0] = xp[0]; xv[1] = xp[1]; xv[2] = xp[2];

    int offv[18];
#pragma unroll
    for (int r = 0; r < 6; ++r) {
      const int   res   = kRes[r];
      const float scale = (float)(res - 1);
      const float step  = 1.0f / (float)(res - 1);
#pragma unroll
      for (int c = 0; c < 3; ++c) {
        const float xx = xv[c];
        // searchsorted(linspace(0,1,res), xx, 'left') closed form + fixup
        int j = (int)ceilf(xx * scale);
        j = (j < 0) ? 0 : ((j > res) ? res : j);
        if (j > 0 && (float)(j - 1) * step >= xx)  --j;
        else if (j < res && (float)j * step < xx)  ++j;
        const int row  = c * (res + 1) + j;
        const int base = (r < 3) ? (kLdsOff[r] << 2) : 0;  // bytes
        offv[r * 3 + c] = base + (row << 8);               // row * 256 B
      }
    }
    // store this point's record: 9x ds_store_b64 (same-wave LDS is in-order)
#pragma unroll
    for (int k = 0; k < 18; k += 2) {
      iv2 t; t[0] = offv[k]; t[1] = offv[k + 1];
      *reinterpret_cast<iv2*>(&idxBuf[waveId][lane][k]) = t;
    }

    // ---- Phase B: 16 iters; groups gather points pb+i and pb+16+i --------
    char* outBase =
        (char*)out + (((size_t)(uint32_t)(pb + laneSel)) << 8) + l16b;
#pragma unroll
    for (int i = 0; i < 16; ++i) {
      fv4 acc = {0.f, 0.f, 0.f, 0.f};
#pragma unroll
      for (int k2 = 0; k2 < 9; ++k2) {
        // broadcast ds_load_b64: offsets (i*72 + k2*8) fold into DS immediate
        const iv2 o = *reinterpret_cast<const iv2*>(rec + i * 18 + 2 * k2);
#pragma unroll
        for (int h = 0; h < 2; ++h) {
          const int k = 2 * k2 + h;
          fv4 v;
          if (k < 9) {  // LDS table: ds_load_b128, bank-conflict-free
            v = *reinterpret_cast<const fv4*>((const char*)smem +
                                              (o[h] + l16b));
          } else {      // L2-resident table: global_load_b128 (saddr+voff)
            const char* W = (k < 12) ? Wg[0] : (k < 15) ? Wg[1] : Wg[2];
            v = *reinterpret_cast<const fv4*>(W + (uint32_t)(o[h] + l16b));
          }
          acc += v;
        }
      }
      if (!GUARD || (pb + laneSel + i) < npts) {
        __builtin_nontemporal_store(
            acc, reinterpret_cast<fv4*>(outBase + ((size_t)i << 8)));
      }
    }
  };

  int pb = ((int)blockIdx.x * WAVES_PER_BLOCK + waveId) * CHUNK;
  for (; pb + CHUNK <= npts; pb += chunkStride) runChunk(pb, FalseT{});
  if (pb < npts) runChunk(pb, TrueT{});
}

extern "C" void kernel_launch(void* const* d_in, const int* in_sizes, int n_in,
                              void* d_out, int out_size, void* d_ws,
                              size_t ws_size, hipStream_t stream) {
  (void)n_in; (void)out_size; (void)d_ws; (void)ws_size;
  const float* x  = (const float*)d_in[0];
  const float* W0 = (const float*)d_in[1];
  const float* W1 = (const float*)d_in[2];
  const float* W2 = (const float*)d_in[3];
  const float* W3 = (const float*)d_in[4];
  const float* W4 = (const float*)d_in[5];
  const float* W5 = (const float*)d_in[6];
  float* out = (float*)d_out;

  const int npts = in_sizes[0] / 3;
  const int ptsPerBlock = THREADS / 32 * CHUNK;  // 256
  int blocks = (npts + ptsPerBlock - 1) / ptsPerBlock;
  if (blocks > 1024) blocks = 1024;  // grid-stride: amortize TDM staging

  hipLaunchKernelGGL(MultiResEmbedding_kernel, dim3(blocks), dim3(THREADS), 0,
                     stream, x, W0, W1, W2, W3, W4, W5, out, npts);
}